// PyTorchFeatureGrid3D_9466107920633
// MI455X (gfx1250) — compile-verified
//
#include <hip/hip_runtime.h>

#define RES 256
#define BLOCK 256
// 256 points * 12 bytes = 3072 bytes staged per block; 3072/16 = 192 async lanes
#define ASYNC_LANES 192

__global__ __launch_bounds__(BLOCK) void trilerp3d_kernel(
    const float* __restrict__ pts,   // (N,3) query points in [0,1)
    const float* __restrict__ grid,  // (256,256,256,3) features
    float* __restrict__ out,         // (N,3)
    int nPts)
{
    __shared__ __align__(16) float s_pts[BLOCK * 3];

    const int tid = threadIdx.x;
    const int i   = blockIdx.x * BLOCK + tid;

    const long long inBytes       = (long long)nPts * 12ll;
    const long long blockByteBase = (long long)blockIdx.x * (BLOCK * 12ll);
    const bool fullBlock = (blockByteBase + BLOCK * 12ll) <= inBytes;

    // Raw LDS byte offset of s_pts = low 32 bits of the flat shared-aperture
    // address (aperture bits live in addr[63:32]; the ISA takes the LDS address from the low 32 bits).
    const unsigned ldsBase = (unsigned)(unsigned long long)(const void*)(&s_pts[0]);

    if (fullBlock) {
        if (tid < ASYNC_LANES) {
            unsigned ldsDst = ldsBase + (unsigned)(tid * 16);
            unsigned long long gSrc =
                (unsigned long long)(const void*)((const char*)pts + blockByteBase + (long long)tid * 16);
            // CDNA5 async global->LDS copy (ASYNCcnt-tracked), 16B per lane, coalesced
            asm volatile("global_load_async_to_lds_b128 %0, %1, off"
                         :: "v"(ldsDst), "v"(gSrc)
                         : "memory");
        }
        asm volatile("s_wait_asynccnt 0" ::: "memory");
    }
    __syncthreads();

    if (i >= nPts) return;

    float px, py, pz;
    if (fullBlock) {
        // Explicit DS readback: the writer is inline asm the optimizer can't
        // see, so read with real ds_load_b32s (LDS latency, no VMEM path,
        // immune to "never-stored LDS -> undef" folding). 4B loads are legal
        // at the 12B point stride in any LDS alignment mode.
        const unsigned lAddr = ldsBase + (unsigned)(tid * 12);
        asm volatile("ds_load_b32 %0, %3\n\t"
                     "ds_load_b32 %1, %3 offset:4\n\t"
                     "ds_load_b32 %2, %3 offset:8\n\t"
                     "s_wait_dscnt 0x0"
                     : "=&v"(px), "=&v"(py), "=&v"(pz)
                     : "v"(lAddr)
                     : "memory");
    } else {
        const float* p = pts + (long long)i * 3;
        px = __builtin_nontemporal_load(p + 0);
        py = __builtin_nontemporal_load(p + 1);
        pz = __builtin_nontemporal_load(p + 2);
    }

    // scaled = p * (R-1); lower = clamp(floor(scaled), 0, R-2); t = scaled - lower
    const float sx = px * 255.0f;
    const float sy = py * 255.0f;
    const float sz = pz * 255.0f;

    int xl = (int)floorf(sx); xl = xl < 0 ? 0 : (xl > 254 ? 254 : xl);
    int yl = (int)floorf(sy); yl = yl < 0 ? 0 : (yl > 254 ? 254 : yl);
    int zl = (int)floorf(sz); zl = zl < 0 ? 0 : (zl > 254 ? 254 : zl);

    const float tx = sx - (float)xl;
    const float ty = sy - (float)yl;
    const float tz = sz - (float)zl;

    // Element (float) offsets into grid[z][y][x][c]; max < 2^26 so 32-bit math is safe.
    const unsigned rowStride   = RES * 3u;        // 768 floats per y-row
    const unsigned sliceStride = RES * RES * 3u;  // 196608 floats per z-slice
    const unsigned xoff = (unsigned)xl * 3u;

    const unsigned zy00 = (unsigned)zl * sliceStride + (unsigned)yl * rowStride + xoff;
    const unsigned zy01 = zy00 + rowStride;                 // (zl, yu)
    const unsigned zy10 = zy00 + sliceStride;               // (zu, yl)
    const unsigned zy11 = zy10 + rowStride;                 // (zu, yu)

    // Each row-pair fetch: 6 contiguous floats = corners (x_l) and (x_u).
    const float* p00 = grid + zy00;
    const float* p01 = grid + zy01;
    const float* p10 = grid + zy10;
    const float* p11 = grid + zy11;

    // Issue all 24 gather dwords up front (compiler batches LOADcnt waits and
    // vectorizes each 6-dword run into wide global loads).
    const float a0 = p00[0], a1 = p00[1], a2 = p00[2], a3 = p00[3], a4 = p00[4], a5 = p00[5];
    const float b0 = p01[0], b1 = p01[1], b2 = p01[2], b3 = p01[3], b4 = p01[4], b5 = p01[5];
    const float c0 = p10[0], c1 = p10[1], c2 = p10[2], c3 = p10[3], c4 = p10[4], c5 = p10[5];
    const float d0 = p11[0], d1 = p11[1], d2 = p11[2], d3 = p11[3], d4 = p11[4], d5 = p11[5];

    const float wx0 = 1.0f - tx;
    const float wy0 = 1.0f - ty;
    const float wz0 = 1.0f - tz;

    // x-lerp (matches reference: c*(1-t) + c'*t)
    const float c00x = a0 * wx0 + a3 * tx;
    const float c00y = a1 * wx0 + a4 * tx;
    const float c00z = a2 * wx0 + a5 * tx;

    const float c01x = b0 * wx0 + b3 * tx;
    const float c01y = b1 * wx0 + b4 * tx;
    const float c01z = b2 * wx0 + b5 * tx;

    const float c10x = c0 * wx0 + c3 * tx;
    const float c10y = c1 * wx0 + c4 * tx;
    const float c10z = c2 * wx0 + c5 * tx;

    const float c11x = d0 * wx0 + d3 * tx;
    const float c11y = d1 * wx0 + d4 * tx;
    const float c11z = d2 * wx0 + d5 * tx;

    // y-lerp
    const float e0x = c00x * wy0 + c01x * ty;
    const float e0y = c00y * wy0 + c01y * ty;
    const float e0z = c00z * wy0 + c01z * ty;

    const float e1x = c10x * wy0 + c11x * ty;
    const float e1y = c10y * wy0 + c11y * ty;
    const float e1z = c10z * wy0 + c11z * ty;

    // z-lerp
    const float rx = e0x * wz0 + e1x * tz;
    const float ry = e0y * wz0 + e1y * tz;
    const float rz = e0z * wz0 + e1z * tz;

    // Non-temporal store: don't let the 25 MB output stream evict grid lines
    // from the 192 MB L2 (grid is ~201 MB and nearly L2-resident across replays).
    float* o = out + (long long)i * 3;
    __builtin_nontemporal_store(rx, o + 0);
    __builtin_nontemporal_store(ry, o + 1);
    __builtin_nontemporal_store(rz, o + 2);
}

extern "C" void kernel_launch(void* const* d_in, const int* in_sizes, int n_in,
                              void* d_out, int out_size, void* d_ws, size_t ws_size,
                              hipStream_t stream) {
    const float* pts  = (const float*)d_in[0];  // (N,3) float32
    const float* grid = (const float*)d_in[1];  // (256,256,256,3) float32
    float* out = (float*)d_out;

    const int nPts = in_sizes[0] / 3;
    const int blocks = (nPts + BLOCK - 1) / BLOCK;

    trilerp3d_kernel<<<blocks, BLOCK, 0, stream>>>(pts, grid, out, nPts);
}